// RevolutionNaive_42365557408277
// MI455X (gfx1250) — compile-verified
//
#include <hip/hip_runtime.h>
#include <hip/hip_bf16.h>

typedef float v2f __attribute__((ext_vector_type(2)));
typedef float v8f __attribute__((ext_vector_type(8)));

#define KS   3
#define C    32
#define Wd   128
#define Hd   128
#define OH   64
#define OW   64
#define IC   201          // K*C*2 + K*K
#define ICP  204          // feat rows padded (201..203 zero) for K=4 WMMA steps
#define OC   1152         // C * NEW*NEW * K*K
#define NT   32           // pixel tile (two WMMA N tiles per wave)
#define NTW  2            // N tiles per wave
#define BLK  256          // 8 waves
#define MPW  144          // M rows per wave (9 tiles of 16); 8*144 = 1152

// LDS layout (floats):
//   p_lds : C*9*NT  =  9216   patches [c][p][n]
//   f_lds : ICP*NT  =  6528   feat    [k][n]   (rows 201..203 zero)
//   sc/sh : OC each =  2304   fused BN scale/shift
//   y_lds : OC*NT   = 36864   post-BN/ReLU conv output [m][n]
#define LDS_FLOATS (C*9*NT + ICP*NT + 2*OC + OC*NT)   // 54912 fl = ~214.5 KB

extern "C" __global__ __launch_bounds__(BLK)
void revolution_fused(const float* __restrict__ x,
                      const float* __restrict__ wmat,   // [OC][IC] row-major
                      const float* __restrict__ gamma,
                      const float* __restrict__ beta,
                      const float* __restrict__ mean,
                      const float* __restrict__ var,
                      float* __restrict__ out)
{
    extern __shared__ float lds[];
    float* p_lds  = lds;                    // 9216
    float* f_lds  = p_lds + C * 9 * NT;     // 6528
    float* sc_lds = f_lds + ICP * NT;       // 1152
    float* sh_lds = sc_lds + OC;            // 1152
    float* y_lds  = sh_lds + OC;            // 36864

    const int tid = threadIdx.x;
    const int blk = blockIdx.x;
    const int b   = blk >> 7;               // 128 pixel tiles per batch image
    const int n0  = (blk & 127) * NT;

    // ---------------- stage 1: patch tile + BN affine into LDS ----------------
    const float* xb = x + (size_t)b * C * Wd * Hd;
    for (int e = tid; e < C * 9 * NT; e += BLK) {
        int n  = e & (NT - 1);
        int cp = e >> 5;                    // c*9 + p
        int c  = cp / 9;
        int p  = cp - c * 9;
        int di = p / 3, dj = p - di * 3;
        int pix = n0 + n;
        int sh  = pix >> 6, sw = pix & 63;
        int h = sh * 2 + di - 1;            // pad = 1, stride = 2
        int w = sw * 2 + dj - 1;
        float v = 0.f;
        if ((unsigned)h < (unsigned)Wd && (unsigned)w < (unsigned)Hd)
            v = xb[(c * Wd + h) * Hd + w];
        p_lds[e] = v;
    }
    for (int m = tid; m < OC; m += BLK) {
        float s = gamma[m] * rsqrtf(var[m] + 1e-5f);
        sc_lds[m] = s;
        sh_lds[m] = beta[m] - mean[m] * s;
    }
    __syncthreads();

    // ---------------- stage 2: feat tile [201(+3 pad)][NT] ----------------
    for (int e = tid; e < ICP * NT; e += BLK) {
        int n = e & (NT - 1);
        int k = e >> 5;
        float v;
        if (k < 9) {                                   // x1: max over channels
            float m = p_lds[k * NT + n];
            #pragma unroll 4
            for (int c = 1; c < C; ++c)
                m = fmaxf(m, p_lds[(c * 9 + k) * NT + n]);
            v = m;
        } else if (k < 9 + C * KS) {                   // x2: max over kernel rows
            int idx = k - 9;
            int c = idx / 3, dj = idx - c * 3;
            int base = (c * 9 + dj) * NT + n;
            v = fmaxf(fmaxf(p_lds[base], p_lds[base + 3 * NT]),
                      p_lds[base + 6 * NT]);
        } else if (k < IC) {                           // x3: max over kernel cols
            int idx = k - (9 + C * KS);
            int c = idx / 3, di = idx - c * 3;
            int base = (c * 9 + di * 3) * NT + n;
            v = fmaxf(fmaxf(p_lds[base], p_lds[base + NT]),
                      p_lds[base + 2 * NT]);
        } else {
            v = 0.f;                                   // K padding rows (201..203)
        }
        f_lds[k * NT + n] = v;
    }
    __syncthreads();

    // ---------------- stage 3: GEMM via V_WMMA_F32_16X16X4_F32 ----------------
    const int wave  = tid >> 5;
    const int lane  = tid & 31;
    const int half  = lane >> 4;     // K/M half selector per ISA layouts
    const int l16   = lane & 15;
    const int mbase = wave * MPW;

    // single A base pointer; per-tile offsets become 24-bit immediates (t*12864B)
    const float* wbase = wmat + (size_t)(mbase + l16) * IC;

    v8f acc[NTW][9];
    #pragma unroll
    for (int nt = 0; nt < NTW; ++nt)
        #pragma unroll
        for (int t = 0; t < 9; ++t)
            #pragma unroll
            for (int v = 0; v < 8; ++v) acc[nt][t][v] = 0.f;

    auto loadA = [&](int kb, v2f* a) {
        const int k0 = kb + 2 * half;
        #pragma unroll
        for (int t = 0; t < 9; ++t) {      // unconditional: k0+1 <= 199 here
            a[t].x = wbase[t * 16 * IC + k0];
            a[t].y = wbase[t * 16 * IC + k0 + 1];
        }
    };
    auto loadB = [&](int kb, v2f* bv) {
        const int k0 = kb + 2 * half;
        #pragma unroll
        for (int nt = 0; nt < NTW; ++nt) {
            bv[nt].x = f_lds[(k0    ) * NT + nt * 16 + l16];
            bv[nt].y = f_lds[(k0 + 1) * NT + nt * 16 + l16];
        }
    };
    auto domac = [&](const v2f* a, const v2f* bv) {
        #pragma unroll
        for (int t = 0; t < 9; ++t) {
            acc[0][t] = __builtin_amdgcn_wmma_f32_16x16x4_f32(
                            false, a[t], false, bv[0], (short)0,
                            acc[0][t], false, false);
            acc[1][t] = __builtin_amdgcn_wmma_f32_16x16x4_f32(
                            false, a[t], false, bv[1], (short)0,
                            acc[1][t], false, false);
        }
    };

    // ping-pong software pipeline (no rotate copies): 50 full k-steps
    v2f a0[9], a1[9], b0[NTW], b1[NTW];
    loadA(0, a0);
    loadB(0, b0);
    #pragma unroll 1
    for (int kb = 0; kb < 192; kb += 8) {   // 24 iterations, steps 0..188
        loadA(kb + 4, a1);
        loadB(kb + 4, b1);
        domac(a0, b0);
        loadA(kb + 8, a0);                  // max kb+8 = 192 (k <= 195)
        loadB(kb + 8, b0);
        domac(a1, b1);
    }
    loadA(196, a1);                         // k 196..199
    loadB(196, b1);
    domac(a0, b0);                          // step kb = 192
    domac(a1, b1);                          // step kb = 196
    // tail: kb = 200 -> only k=200 is real. A cols 201..203 selected to zero
    // (branch-free cndmask); B rows 201..203 are zero in f_lds.
    {
        const int k0 = 200 + 2 * half;
        v2f bt[NTW];
        #pragma unroll
        for (int nt = 0; nt < NTW; ++nt) {
            bt[nt].x = f_lds[(k0    ) * NT + nt * 16 + l16];
            bt[nt].y = f_lds[(k0 + 1) * NT + nt * 16 + l16];
        }
        #pragma unroll
        for (int t = 0; t < 9; ++t) {
            float w200 = wbase[t * 16 * IC + 200];    // in bounds for every row
            v2f at;
            at.x = half ? 0.f : w200;                 // col 202 -> 0 on upper half
            at.y = 0.f;                               // cols 201/203 -> 0
            acc[0][t] = __builtin_amdgcn_wmma_f32_16x16x4_f32(
                            false, at, false, bt[0], (short)0,
                            acc[0][t], false, false);
            acc[1][t] = __builtin_amdgcn_wmma_f32_16x16x4_f32(
                            false, at, false, bt[1], (short)0,
                            acc[1][t], false, false);
        }
    }

    // ---------------- stage 4: BN + ReLU, stage y tile in LDS ----------------
    #pragma unroll
    for (int t = 0; t < 9; ++t) {
        #pragma unroll
        for (int v = 0; v < 8; ++v) {
            int m = mbase + t * 16 + 8 * half + v;     // C layout: M = v + 8*half
            float s = sc_lds[m], h = sh_lds[m];
            #pragma unroll
            for (int nt = 0; nt < NTW; ++nt) {
                float yv = acc[nt][t][v] * s + h;
                y_lds[m * NT + nt * 16 + l16] = fmaxf(yv, 0.f);
            }
        }
    }
    __syncthreads();

    // ---------------- stage 5: dynamic filtering + scatter to output ----------
    float* outb = out + (size_t)b * C * Wd * Hd;
    for (int idx = tid; idx < C * 4 * NT; idx += BLK) {
        int n   = idx & (NT - 1);
        int rem = idx >> 5;              // c*4 + e
        int c   = rem >> 2;
        int e   = rem & 3;               // e = i*2 + j
        float s = 0.f;
        #pragma unroll
        for (int p = 0; p < 9; ++p) {
            float pv = p_lds[(c * 9 + p) * NT + n];
            float yv = y_lds[(c * 36 + p * 4 + e) * NT + n];
            s += pv * yv;
        }
        s *= (1.f / 9.f);
        int pix = n0 + n;
        int sh = pix >> 6, sw = pix & 63;
        int i = e >> 1, j = e & 1;
        outb[(c * Wd + (sh * 2 + i)) * Hd + (sw * 2 + j)] = s;
    }
}

extern "C" void kernel_launch(void* const* d_in, const int* in_sizes, int n_in,
                              void* d_out, int out_size, void* d_ws, size_t ws_size,
                              hipStream_t stream) {
    const float* x    = (const float*)d_in[0];
    const float* wmat = (const float*)d_in[1];
    const float* gam  = (const float*)d_in[2];
    const float* bet  = (const float*)d_in[3];
    const float* mean = (const float*)d_in[4];
    const float* var  = (const float*)d_in[5];
    float*       out  = (float*)d_out;

    const size_t lds_bytes = (size_t)LDS_FLOATS * sizeof(float); // ~214.5 KB < 320 KB/WGP
    dim3 grid(8 * (OH * OW / NT));   // 8 batches * 128 pixel tiles = 1024 WGs
    revolution_fused<<<grid, BLK, lds_bytes, stream>>>(x, wmat, gam, bet, mean, var, out);
}